// KGraph_88811333747382
// MI455X (gfx1250) — compile-verified
//
#include <hip/hip_runtime.h>
#include <hip/hip_bf16.h>
#include <math.h>

// ---------------------------------------------------------------------------
// kNN graph + local covariance eigendecomposition for MI455X (gfx1250).
//
// Distance tiles via V_WMMA_F32_16X16X4_F32:
//   A (16x4 f32) = query features scaled by -2 (K=3 padded to 4)
//   B (4x16 f32) = candidate features (padded)
//   C (16x16 f32) preloaded with sq_c[n]  => D[m][n] = sq_c[n] - 2*dot(q_m,c_n)
//   dist = sq_q[m] + D[m][n]
//
// VGPR layouts per CDNA5 ISA (CDNA5 ISA, 32-bit matrices):
//   A 16x4 : lanes 0-15 hold (M=lane, K=0/1); lanes 16-31 hold (M=lane-16, K=2/3)
//   B 4x16 : lanes 0-15 hold (K=0/1, N=lane); lanes 16-31 hold (K=2/3, N=lane-16)
//   C/D    : VGPR r -> (M=r, N=lane) lanes 0-15 ; (M=r+8, N=lane-16) lanes 16-31
// ---------------------------------------------------------------------------

typedef __attribute__((ext_vector_type(2))) float v2f;
typedef __attribute__((ext_vector_type(8))) float v8f;

#define KNBR 16
#define ROWS 20   // LDS row stride in floats: 80 B, 16 B aligned, conflict-free

// ---------------------------------------------------------------- sq norms --
__global__ void sqnorm_kernel(const float* __restrict__ feat,
                              float* __restrict__ sq, int n) {
    int i = blockIdx.x * blockDim.x + threadIdx.x;
    if (i < n) {
        float x = feat[3 * i + 0];
        float y = feat[3 * i + 1];
        float z = feat[3 * i + 2];
        sq[i] = x * x + y * y + z * z;
    }
}

// ------------------------------------------------------------- WMMA kNN ----
// One wave per 32 queries; loops over candidates of the same cloud in 16-wide
// tiles; 2 WMMAs per tile; software-pipelined candidate loads.
__global__ __launch_bounds__(32)
void knn_wmma_kernel(const float* __restrict__ feat,   // [B*N, 3]
                     const float* __restrict__ sq,     // [B*N]
                     int* __restrict__ sid,            // [B*N*K] global src ids
                     int* __restrict__ tid,            // [B*N*K] global tgt ids
                     int n)                            // points per cloud
{
    __shared__ __align__(16) float lds[2 * 16 * ROWS]; // two padded 16x16 tiles

    const int lane = threadIdx.x;                      // 0..31, wave32
    const int l16  = lane & 15;
    const int hsel = lane >> 4;                        // 0: K=0/1 half, 1: K=2/3

    const int tilesPerBatch = n / 32;
    const int b     = blockIdx.x / tilesPerBatch;
    const int qbase = b * n + (blockIdx.x % tilesPerBatch) * 32;

    // ---- A0 (queries qbase..+15) / A1 (+16..+31), scaled by -2, branch-free
    const int r0 = qbase + l16;
    const int r1 = r0 + 16;
    {
        float f00 = feat[3 * r0 + 2 * hsel];   // x (hsel=0) or z (hsel=1)
        float f01 = feat[3 * r0 + 1];          // y
        float f10 = feat[3 * r1 + 2 * hsel];
        float f11 = feat[3 * r1 + 1];
        // use below
        (void)f01; (void)f11;
        // stored into a0/a1 after declaration
    }
    v2f a0, a1;
    {
        float f00 = feat[3 * r0 + 2 * hsel];
        float f01 = feat[3 * r0 + 1];
        float f10 = feat[3 * r1 + 2 * hsel];
        float f11 = feat[3 * r1 + 1];
        a0[0] = -2.0f * f00;  a0[1] = hsel ? 0.0f : (-2.0f * f01);
        a1[0] = -2.0f * f10;  a1[1] = hsel ? 0.0f : (-2.0f * f11);
    }

    // lane owns one query row for the selection phase
    const int   q   = qbase + lane;
    const float sqq = sq[q];
    const float* myrow = &lds[hsel * (16 * ROWS) + l16 * ROWS];

    // register-resident sorted top-16 (ascending by (dist, idx))
    float dbest[KNBR];
    int   ibest[KNBR];
#pragma unroll
    for (int t = 0; t < KNBR; ++t) { dbest[t] = 3.402823466e38f; ibest[t] = 0x7fffffff; }

    const int rowAdd = hsel ? 8 : 0;                   // C/D row mapping

    // ---- software-pipeline prologue: load candidate tile 0 ----
    float nfa, nfb, nsqc;
    {
        const int cr = b * n + l16;
        nfa  = feat[3 * cr + 2 * hsel];
        nfb  = feat[3 * cr + 1];
        nsqc = sq[cr];
    }

    for (int cblk = 0; cblk < n; cblk += 16) {
        const float fa = nfa, fb = nfb, sqc = nsqc;

        // issue next tile's loads early (wrap to 0: unconditional, in-bounds)
        {
            int nextblk = cblk + 16;
            nextblk = (nextblk < n) ? nextblk : 0;
            const int ncr = b * n + nextblk + l16;
            nfa  = feat[3 * ncr + 2 * hsel];
            nfb  = feat[3 * ncr + 1];
            nsqc = sq[ncr];
        }

        // ---- B tile (branch-free lane layout) ----
        v2f bb;
        bb[0] = fa;
        bb[1] = hsel ? 0.0f : fb;

        // ---- C accumulator preloaded with sq_c[n] (column broadcast) ----
        v8f cacc;
#pragma unroll
        for (int r = 0; r < 8; ++r) cacc[r] = sqc;

        // ---- two f32 WMMAs: D = (-2*Q) x C^T + sq_c ----
        v8f d0 = __builtin_amdgcn_wmma_f32_16x16x4_f32(
            false, a0, false, bb, (short)0, cacc, false, false);
        v8f d1 = __builtin_amdgcn_wmma_f32_16x16x4_f32(
            false, a1, false, bb, (short)0, cacc, false, false);

        // ---- stage both 16x16 tiles through LDS ----
#pragma unroll
        for (int r = 0; r < 8; ++r)
            lds[(r + rowAdd) * ROWS + l16] = d0[r];
#pragma unroll
        for (int r = 0; r < 8; ++r)
            lds[16 * ROWS + (r + rowAdd) * ROWS + l16] = d1[r];
        __syncthreads();

        // ---- fetch my 16 distances as 4x ds_load_b128 ----
        float buf[16];
        {
            const float4* rp = (const float4*)myrow;
            float4 p0 = rp[0], p1 = rp[1], p2 = rp[2], p3 = rp[3];
            buf[0] = p0.x; buf[1] = p0.y; buf[2]  = p0.z; buf[3]  = p0.w;
            buf[4] = p1.x; buf[5] = p1.y; buf[6]  = p1.z; buf[7]  = p1.w;
            buf[8] = p2.x; buf[9] = p2.y; buf[10] = p2.z; buf[11] = p2.w;
            buf[12] = p3.x; buf[13] = p3.y; buf[14] = p3.z; buf[15] = p3.w;
        }
        __syncthreads();                               // LDS free for next tile

        // ---- branch-free top-16 maintenance ----
#pragma unroll
        for (int j = 0; j < 16; ++j) {
            float d  = sqq + buf[j];
            int   ci = cblk + j;                       // local index in cloud
            // single early-reject branch; bitwise combine to avoid exec churn
            int bet = (int)(d < dbest[KNBR - 1]) |
                      ((int)(d == dbest[KNBR - 1]) & (int)(ci < ibest[KNBR - 1]));
            if (bet) {
                float cd = d; int cc = ci;
#pragma unroll
                for (int t = 0; t < KNBR; ++t) {
                    int sw = (int)(cd < dbest[t]) |
                             ((int)(cd == dbest[t]) & (int)(cc < ibest[t]));
                    float od = dbest[t]; int oi = ibest[t];
                    dbest[t] = sw ? cd : od;  ibest[t] = sw ? cc : oi;
                    cd       = sw ? od : cd;  cc       = sw ? oi : cc;
                }
            }
        }
    }

    // ---- emit edges (ascending (dist, idx): matches stable top_k(-dist)) ----
#pragma unroll
    for (int t = 0; t < KNBR; ++t) {
        sid[q * KNBR + t] = b * n + ibest[t];
        tid[q * KNBR + t] = q;
    }
}

// ------------------------------------------------- covariance + 3x3 eigh ---
__device__ inline void eigvec3(float a00, float a01, float a02,
                               float a11, float a12, float a22,
                               float l, float v[3]) {
    float r0x = a00 - l, r0y = a01,     r0z = a02;
    float r1x = a01,     r1y = a11 - l, r1z = a12;
    float r2x = a02,     r2y = a12,     r2z = a22 - l;
    float cx0 = r0y * r1z - r0z * r1y, cy0 = r0z * r1x - r0x * r1z, cz0 = r0x * r1y - r0y * r1x;
    float cx1 = r0y * r2z - r0z * r2y, cy1 = r0z * r2x - r0x * r2z, cz1 = r0x * r2y - r0y * r2x;
    float cx2 = r1y * r2z - r1z * r2y, cy2 = r1z * r2x - r1x * r2z, cz2 = r1x * r2y - r1y * r2x;
    float n0 = cx0 * cx0 + cy0 * cy0 + cz0 * cz0;
    float n1 = cx1 * cx1 + cy1 * cy1 + cz1 * cz1;
    float n2 = cx2 * cx2 + cy2 * cy2 + cz2 * cz2;
    float nx = cx0, ny = cy0, nz = cz0, nn = n0;
    if (n1 > nn) { nx = cx1; ny = cy1; nz = cz1; nn = n1; }
    if (n2 > nn) { nx = cx2; ny = cy2; nz = cz2; nn = n2; }
    if (nn < 1e-30f) { v[0] = 1.0f; v[1] = 0.0f; v[2] = 0.0f; return; }
    float inv = rsqrtf(nn);
    v[0] = nx * inv; v[1] = ny * inv; v[2] = nz * inv;
}

__global__ void cov_eig_kernel(const float* __restrict__ pos,   // [B*N,3]
                               const int* __restrict__ sid,     // [B*N*K] global ids
                               float* __restrict__ eig_out,     // [B*N,3]
                               float* __restrict__ vec_out,     // [B*N,3,3]
                               int total) {
    int i = blockIdx.x * blockDim.x + threadIdx.x;
    if (i >= total) return;

    float px = pos[3 * i + 0], py = pos[3 * i + 1], pz = pos[3 * i + 2];
    float a00 = 0, a01 = 0, a02 = 0, a11 = 0, a12 = 0, a22 = 0;
#pragma unroll
    for (int t = 0; t < KNBR; ++t) {
        int s = sid[i * KNBR + t];
        float dx = pos[3 * s + 0] - px;
        float dy = pos[3 * s + 1] - py;
        float dz = pos[3 * s + 2] - pz;
        a00 += dx * dx; a01 += dx * dy; a02 += dx * dz;
        a11 += dy * dy; a12 += dy * dz; a22 += dz * dz;
    }

    // Smith's closed-form symmetric 3x3 eigenvalues (ascending like eigh).
    float qm  = (a00 + a11 + a22) * (1.0f / 3.0f);
    float p1  = a01 * a01 + a02 * a02 + a12 * a12;
    float b00 = a00 - qm, b11 = a11 - qm, b22 = a22 - qm;
    float p2  = b00 * b00 + b11 * b11 + b22 * b22 + 2.0f * p1;

    float e0, e1, e2;
    float v0[3], v1[3], v2v[3];
    if (p2 <= 1e-30f) {
        e0 = e1 = e2 = qm;
        v0[0] = 1; v0[1] = 0; v0[2] = 0;
        v1[0] = 0; v1[1] = 1; v1[2] = 0;
        v2v[0] = 0; v2v[1] = 0; v2v[2] = 1;
    } else {
        float p  = sqrtf(p2 * (1.0f / 6.0f));
        float ip = 1.0f / p;
        float c00 = b00 * ip, c11 = b11 * ip, c22 = b22 * ip;
        float c01 = a01 * ip, c02 = a02 * ip, c12 = a12 * ip;
        float detB = c00 * (c11 * c22 - c12 * c12)
                   - c01 * (c01 * c22 - c12 * c02)
                   + c02 * (c01 * c12 - c11 * c02);
        float r = detB * 0.5f;
        r = fminf(1.0f, fmaxf(-1.0f, r));
        float phi = acosf(r) * (1.0f / 3.0f);
        float big   = qm + 2.0f * p * cosf(phi);
        float small = qm + 2.0f * p * cosf(phi + 2.0943951023931953f); // +2pi/3
        float mid   = 3.0f * qm - big - small;
        e0 = small; e1 = mid; e2 = big;

        eigvec3(a00, a01, a02, a11, a12, a22, e0, v0);
        eigvec3(a00, a01, a02, a11, a12, a22, e2, v2v);
        // middle vector = cross(v2, v0), enforcing orthogonality
        v1[0] = v2v[1] * v0[2] - v2v[2] * v0[1];
        v1[1] = v2v[2] * v0[0] - v2v[0] * v0[2];
        v1[2] = v2v[0] * v0[1] - v2v[1] * v0[0];
        float nn = v1[0] * v1[0] + v1[1] * v1[1] + v1[2] * v1[2];
        if (nn < 1e-30f) {
            // v0 ~ +-v2 (degenerate); build any vector orthogonal to v2
            float ax = fabsf(v2v[0]), ay = fabsf(v2v[1]), az = fabsf(v2v[2]);
            float ux, uy, uz;
            if (ax <= ay && ax <= az) { ux = 1; uy = 0; uz = 0; }
            else if (ay <= az)        { ux = 0; uy = 1; uz = 0; }
            else                      { ux = 0; uy = 0; uz = 1; }
            v1[0] = v2v[1] * uz - v2v[2] * uy;
            v1[1] = v2v[2] * ux - v2v[0] * uz;
            v1[2] = v2v[0] * uy - v2v[1] * ux;
            nn = v1[0] * v1[0] + v1[1] * v1[1] + v1[2] * v1[2];
        }
        float inv = rsqrtf(fmaxf(nn, 1e-38f));
        v1[0] *= inv; v1[1] *= inv; v1[2] *= inv;
    }

    eig_out[3 * i + 0] = e0;
    eig_out[3 * i + 1] = e1;
    eig_out[3 * i + 2] = e2;
    // eigh convention: columns are eigenvectors -> vec[r][c] = c-th eigenvector
    float* V = &vec_out[9 * i];
    V[0] = v0[0]; V[1] = v1[0]; V[2] = v2v[0];
    V[3] = v0[1]; V[4] = v1[1]; V[5] = v2v[1];
    V[6] = v0[2]; V[7] = v1[2]; V[8] = v2v[2];
}

// ---------------------------------------------------------------- launch ---
extern "C" void kernel_launch(void* const* d_in, const int* in_sizes, int n_in,
                              void* d_out, int out_size, void* d_ws, size_t ws_size,
                              hipStream_t stream) {
    const float* pos = (const float*)d_in[0];          // [B, 8192, 3] f32
    const int total = in_sizes[0] / 3;                 // B*N = 16384
    const int N = 8192;                                // points per cloud

    // d_out layout (reference return order, flat):
    //   sid_euc, tid_euc, sid_eig, tid_eig : int32  [total*K] each
    //   eig : f32 [total*3] ; vec : f32 [total*9]
    int*   sid_euc = (int*)d_out;
    int*   tid_euc = sid_euc + (size_t)total * KNBR;
    int*   sid_eig = tid_euc + (size_t)total * KNBR;
    int*   tid_eig = sid_eig + (size_t)total * KNBR;
    float* eig     = (float*)(tid_eig + (size_t)total * KNBR);
    float* vec     = eig + (size_t)total * 3;

    float* sqbuf = (float*)d_ws;                       // total floats of scratch

    const int blocksPts = (total + 255) / 256;
    const int blocksKnn = total / 32;                  // one wave per 32 queries

    // Pass 1: kNN in Euclidean space
    sqnorm_kernel<<<blocksPts, 256, 0, stream>>>(pos, sqbuf, total);
    knn_wmma_kernel<<<blocksKnn, 32, 0, stream>>>(pos, sqbuf, sid_euc, tid_euc, N);

    // Covariance of neighbor offsets + 3x3 symmetric eigh
    cov_eig_kernel<<<(total + 127) / 128, 128, 0, stream>>>(pos, sid_euc, eig, vec, total);

    // Pass 2: kNN in eigenvalue feature space (reads eig straight from d_out)
    sqnorm_kernel<<<blocksPts, 256, 0, stream>>>(eig, sqbuf, total);
    knn_wmma_kernel<<<blocksKnn, 32, 0, stream>>>(eig, sqbuf, sid_eig, tid_eig, N);
}